// PredictionHead_65996467470713
// MI455X (gfx1250) — compile-verified
//
#include <hip/hip_runtime.h>
#include <hip/hip_bf16.h>
#include <cstdint>

// QA span prediction head, collapsed to log-domain sliding-window max:
//   start[b] = argmax_i ( s[b,i] + max_{j in [i, min(i+W,L-1)]} e[b,j] )
//   end[b]   = argmax_j ( e[b,j] + max_{i in [max(0,j-W), j]} s[b,i] )
// Exact w.r.t. the softmax/outer-product reference (exp monotone, Z constant per row).

#define QA_L 3072
#define QA_THREADS 256

__global__ __launch_bounds__(QA_THREADS) void qa_span_argmax_kernel(
    const float* __restrict__ start_logits,
    const float* __restrict__ end_logits,
    const int*   __restrict__ answer_limit,
    int*         __restrict__ out,   // [2*B]: start pointers, then end pointers
    int B)
{
  __shared__ __align__(16) float ls[QA_L];
  __shared__ __align__(16) float le[QA_L];
  __shared__ float rv[QA_THREADS];
  __shared__ int   ri[QA_THREADS];

  const int b   = blockIdx.x;
  const int tid = threadIdx.x;
  const int W   = answer_limit[0];

  // LDS byte offsets (low 32 bits of the flat shared address == LDS offset).
  const unsigned lds_s = (unsigned)(uintptr_t)(void*)ls;
  const unsigned lds_e = (unsigned)(uintptr_t)(void*)le;
  const unsigned long long g_s =
      (unsigned long long)(uintptr_t)(const void*)(start_logits + (size_t)b * QA_L);
  const unsigned long long g_e =
      (unsigned long long)(uintptr_t)(const void*)(end_logits + (size_t)b * QA_L);

  // Stage both rows (12 KB each) into LDS via CDNA5 async DMA (ASYNCcnt).
  const int BYTES = QA_L * 4;
  for (int off = tid * 16; off < BYTES; off += QA_THREADS * 16) {
    unsigned           ds = lds_s + (unsigned)off;
    unsigned           de = lds_e + (unsigned)off;
    unsigned long long as = g_s + (unsigned long long)off;
    unsigned long long ae = g_e + (unsigned long long)off;
    asm volatile("global_load_async_to_lds_b128 %0, %1, off"
                 :: "v"(ds), "v"(as) : "memory");
    asm volatile("global_load_async_to_lds_b128 %0, %1, off"
                 :: "v"(de), "v"(ae) : "memory");
  }
  asm volatile("s_wait_asynccnt 0" ::: "memory");
  __syncthreads();

  // ---- start pointer: argmax_i ( ls[i] + max_{j in [i, i+W]} le[j] ) ----
  float bestv = -3.402823466e38f;
  int   besti = 0x7fffffff;
  for (int i = tid; i < QA_L; i += QA_THREADS) {
    const int jmax = min(i + W, QA_L - 1);
    float wm = le[i];
    for (int j = i + 1; j <= jmax; ++j) wm = fmaxf(wm, le[j]);
    const float v = ls[i] + wm;
    if (v > bestv) { bestv = v; besti = i; }  // strict > keeps first index
  }
  rv[tid] = bestv; ri[tid] = besti;
  __syncthreads();
  for (int s = QA_THREADS / 2; s > 0; s >>= 1) {
    if (tid < s) {
      const float ov = rv[tid + s];
      const int   oi = ri[tid + s];
      if (ov > rv[tid] || (ov == rv[tid] && oi < ri[tid])) { rv[tid] = ov; ri[tid] = oi; }
    }
    __syncthreads();
  }
  if (tid == 0) out[b] = ri[0];
  __syncthreads();

  // ---- end pointer: argmax_j ( le[j] + max_{i in [j-W, j]} ls[i] ) ----
  bestv = -3.402823466e38f;
  besti = 0x7fffffff;
  for (int j = tid; j < QA_L; j += QA_THREADS) {
    const int imin = max(j - W, 0);
    float wm = ls[j];
    for (int i = j - 1; i >= imin; --i) wm = fmaxf(wm, ls[i]);
    const float v = le[j] + wm;
    if (v > bestv) { bestv = v; besti = j; }
  }
  rv[tid] = bestv; ri[tid] = besti;
  __syncthreads();
  for (int s = QA_THREADS / 2; s > 0; s >>= 1) {
    if (tid < s) {
      const float ov = rv[tid + s];
      const int   oi = ri[tid + s];
      if (ov > rv[tid] || (ov == rv[tid] && oi < ri[tid])) { rv[tid] = ov; ri[tid] = oi; }
    }
    __syncthreads();
  }
  if (tid == 0) out[B + b] = ri[0];
}

extern "C" void kernel_launch(void* const* d_in, const int* in_sizes, int n_in,
                              void* d_out, int out_size, void* d_ws, size_t ws_size,
                              hipStream_t stream) {
  (void)in_sizes; (void)n_in; (void)d_ws; (void)ws_size;
  const float* s   = (const float*)d_in[0];
  const float* e   = (const float*)d_in[1];
  const int*   lim = (const int*)d_in[2];
  const int B = out_size / 2;              // 8: tuple (start[B], end[B]) flattened
  int* out = (int*)d_out;                  // reference returns int32 indices
  qa_span_argmax_kernel<<<B, QA_THREADS, 0, stream>>>(s, e, lim, out, B);
}